// QueryAndGroupKP_38044820308018
// MI455X (gfx1250) — compile-verified
//
#include <hip/hip_runtime.h>

typedef __attribute__((ext_vector_type(2))) float v2f;
typedef __attribute__((ext_vector_type(8))) float v8f;

#define BQ_B 4
#define BQ_N 16384
#define BQ_P 1024
#define BQ_C 128
#define BQ_S 64
#define BQ_R2 0.04f          // RADIUS^2
#define BQ_INVR 5.0f         // 1/RADIUS

// -------------------------------------------------------------------------
// Kernel 1: ball query via V_WMMA_F32_16X16X4_F32.
// One wave (32 lanes) owns 16 centers; each iteration one WMMA produces a
// 16x16 tile of squared distances (16 centers x 16 consecutive points).
//   A (16x4, MxK):  row m = (-2cx, -2cy, -2cz, |c|^2)
//   B (4x16, KxN):  col n = (px, py, pz, 1)
//   C (16x16):      C[m][n] = |p_n|^2  (broadcast per lane, n = lane%16)
//   D = A*B + C = ||c_m - p_n||^2
// Per-row in-radius masks come from __ballot over the 8 D VGPRs
// (VGPR v -> row v in bits [15:0], row v+8 in bits [31:16]); ordered
// append (first-64-ascending) via popcount-prefix compaction in SGPRs.
// Point loads are software-pipelined: chunk i+1's global_load_b96 issues
// before chunk i's WMMA+selection work so s_wait_loadcnt lands after ~250
// instructions of cover (only ~4 waves/WGP are resident to hide latency).
// -------------------------------------------------------------------------
__global__ __launch_bounds__(32) void ball_query_wmma(
    const float* __restrict__ xyz,     // (B, N, 3)
    const float* __restrict__ nxyz,    // (B, P, 3)
    int* __restrict__ idx_out)         // (B, P, S)
{
    const int lane = threadIdx.x;          // 0..31 (wave32)
    const int n16  = lane & 15;
    const int half = lane >> 4;
    const int blk  = blockIdx.x;           // b*(P/16) + g
    const int b    = blk / (BQ_P / 16);
    const int g    = blk % (BQ_P / 16);

    // --- build A fragment (constant over the whole scan) ---
    const float* cp = nxyz + ((size_t)b * BQ_P + (size_t)g * 16 + n16) * 3;
    const float cx = cp[0], cy = cp[1], cz = cp[2];
    v2f A;
    A.x = half ? (-2.0f * cz) : (-2.0f * cx);
    A.y = half ? (cx * cx + cy * cy + cz * cz) : (-2.0f * cy);

    const float* xb = xyz + (size_t)b * BQ_N * 3;
    int* rows = idx_out + ((size_t)b * BQ_P + (size_t)g * 16) * BQ_S;

    int cnt[16];
    int first[16];
#pragma unroll
    for (int r = 0; r < 16; ++r) { cnt[r] = 0; first[r] = BQ_N; }

    // --- prologue: load chunk 0 ---
    float px = xb[(size_t)n16 * 3 + 0];
    float py = xb[(size_t)n16 * 3 + 1];
    float pz = xb[(size_t)n16 * 3 + 2];

    for (int base = 0; base < BQ_N; base += 16) {
        // --- issue next chunk's load first (index clamped: branch-free,
        //     last iteration reloads the final chunk harmlessly) ---
        const int nb  = (base + 16 < BQ_N) ? (base + 16) : (BQ_N - 16);
        const int npi = nb + n16;
        const float qx = xb[(size_t)npi * 3 + 0];
        const float qy = xb[(size_t)npi * 3 + 1];
        const float qz = xb[(size_t)npi * 3 + 2];

        // prefetch chunk i+2 of the streaming xyz scan (global_prefetch_b8)
        {
            const int pb = (base + 32 < BQ_N) ? (base + 32) : (BQ_N - 16);
            __builtin_prefetch(xb + (size_t)pb * 3, 0, 1);
        }

        // --- process current chunk: B fragment + C (=|p|^2) ---
        const float pn2 = px * px + py * py + pz * pz;
        v2f Bm;
        Bm.x = half ? pz : px;
        Bm.y = half ? 1.0f : py;
        v8f C = {pn2, pn2, pn2, pn2, pn2, pn2, pn2, pn2};

        // D[m][n] = squared distance  (v_wmma_f32_16x16x4_f32)
        v8f D = __builtin_amdgcn_wmma_f32_16x16x4_f32(
            false, A, false, Bm, (short)0, C, false, false);

        // --- per-row in-radius masks (uniform scalars via ballot) ---
        unsigned bal[8];
#pragma unroll
        for (int v = 0; v < 8; ++v)
            bal[v] = (unsigned)__ballot(D[v] < BQ_R2);

        // --- ordered append for each of the 16 rows ---
#pragma unroll
        for (int r = 0; r < 16; ++r) {
            const unsigned mk = (r < 8) ? (bal[r] & 0xffffu) : (bal[r - 8] >> 16);
            if (cnt[r] < BQ_S && mk) {
                if (cnt[r] == 0) first[r] = base + (__ffs((int)mk) - 1);
                if (lane < 16 && ((mk >> lane) & 1u)) {
                    const int pos = cnt[r] + __popc(mk & ((1u << lane) - 1u));
                    if (pos < BQ_S) rows[(size_t)r * BQ_S + pos] = base + lane;
                }
                cnt[r] += __popc(mk);
            }
        }

        // --- rotate double buffer (wait for next-chunk load lands here) ---
        px = qx; py = qy; pz = qz;

        // --- early exit when all 16 centers are full (uniform branch) ---
        bool done = true;
#pragma unroll
        for (int r = 0; r < 16; ++r) done = done && (cnt[r] >= BQ_S);
        if (done) break;
    }

    // --- pad remaining slots with the first index (or N if none found) ---
#pragma unroll
    for (int r = 0; r < 16; ++r) {
        const int c = cnt[r] > BQ_S ? BQ_S : cnt[r];
        const int fillv = (c > 0) ? first[r] : BQ_N;
        int* row = rows + (size_t)r * BQ_S;
#pragma unroll
        for (int pass = 0; pass < 2; ++pass) {
            const int s = lane + pass * 32;
            if (s >= c) row[s] = fillv;
        }
    }
}

// -------------------------------------------------------------------------
// Kernel 2: gather + write all three outputs. One thread per (b,p,s); s is
// the fastest index so stores are lane-coalesced. This kernel is the
// bandwidth-dominant part (~280 MB total traffic -> ~12-13 us at 23.3 TB/s).
// -------------------------------------------------------------------------
__global__ __launch_bounds__(256) void gather_write(
    const float* __restrict__ xyz,     // (B, N, 3)
    const float* __restrict__ nxyz,    // (B, P, 3)
    const float* __restrict__ feat,    // (B, C, N)
    float* __restrict__ out)
{
    const int t = blockIdx.x * 256 + threadIdx.x;     // 0 .. B*P*S-1
    const int s = t & (BQ_S - 1);
    const int p = (t >> 6) & (BQ_P - 1);
    const int b = t >> 16;

    const size_t PS = (size_t)BQ_P * BQ_S;            // 65536
    const size_t NF = (size_t)BQ_B * (3 + BQ_C) * PS; // new_features elems
    const size_t GX = (size_t)BQ_B * 3 * PS;          // grouped_xyz elems

    const int* idx = (const int*)out + NF + GX;       // written by kernel 1
    const int id  = idx[((size_t)b * BQ_P + p) * BQ_S + s];
    const int fid = idx[((size_t)b * BQ_P + p) * BQ_S];
    // dup -> sentinel (reference: idx = where(dup, N, idx0))
    const int ide = (s > 0 && id == fid) ? BQ_N : id;
    const bool pad = (ide >= BQ_N);

    const float* cp = nxyz + ((size_t)b * BQ_P + p) * 3;
    float g0, g1, g2;
    if (pad) {
        g0 = 1.0e6f - cp[0]; g1 = 1.0e6f - cp[1]; g2 = 1.0e6f - cp[2];
    } else {
        const float* q = xyz + ((size_t)b * BQ_N + ide) * 3;
        g0 = q[0] - cp[0]; g1 = q[1] - cp[1]; g2 = q[2] - cp[2];
    }

    const size_t ps = (size_t)p * BQ_S + s;

    // grouped_xyz (B,3,P,S)
    float* gxo = out + NF + (size_t)b * 3 * PS;
    gxo[0 * PS + ps] = g0;
    gxo[1 * PS + ps] = g1;
    gxo[2 * PS + ps] = g2;

    // new_features channels 0..2 = xyz_feature (pad -> 0, else offset/R)
    float* nfo = out + (size_t)b * (3 + BQ_C) * PS;
    nfo[0 * PS + ps] = pad ? 0.0f : g0 * BQ_INVR;
    nfo[1 * PS + ps] = pad ? 0.0f : g1 * BQ_INVR;
    nfo[2 * PS + ps] = pad ? 0.0f : g2 * BQ_INVR;

    // new_features channels 3..130 = gathered features (pad -> 0)
    float* fo = nfo + 3 * PS;
    if (pad) {
#pragma unroll 4
        for (int c = 0; c < BQ_C; ++c)
            fo[(size_t)c * PS + ps] = 0.0f;
    } else {
        const float* fb = feat + (size_t)b * BQ_C * BQ_N + ide;
#pragma unroll 4
        for (int c = 0; c < BQ_C; ++c)
            fo[(size_t)c * PS + ps] = fb[(size_t)c * BQ_N];
    }
    // idx0 output already in place (written by kernel 1).
}

// -------------------------------------------------------------------------
extern "C" void kernel_launch(void* const* d_in, const int* in_sizes, int n_in,
                              void* d_out, int out_size, void* d_ws, size_t ws_size,
                              hipStream_t stream) {
    (void)in_sizes; (void)n_in; (void)out_size; (void)d_ws; (void)ws_size;
    const float* xyz  = (const float*)d_in[0];   // (4,16384,3)
    const float* nxyz = (const float*)d_in[1];   // (4,1024,3)
    const float* feat = (const float*)d_in[2];   // (4,128,16384)
    float* out = (float*)d_out;

    const size_t PS = (size_t)BQ_P * BQ_S;
    const size_t NF = (size_t)BQ_B * (3 + BQ_C) * PS;
    const size_t GX = (size_t)BQ_B * 3 * PS;
    int* idx = (int*)out + NF + GX;              // idx0 lives at the tail of d_out

    // Kernel 1: B*(P/16) waves, one wave per 16 centers.
    hipLaunchKernelGGL(ball_query_wmma,
                       dim3(BQ_B * (BQ_P / 16)), dim3(32), 0, stream,
                       xyz, nxyz, idx);

    // Kernel 2: one thread per (b,p,s).
    hipLaunchKernelGGL(gather_write,
                       dim3((BQ_B * BQ_P * BQ_S) / 256), dim3(256), 0, stream,
                       xyz, nxyz, feat, out);
}